// GCNEncoder_37022618092193
// MI455X (gfx1250) — compile-verified
//
#include <hip/hip_runtime.h>

#define N_NODES  50000
#define N_EDGES  1600000
#define N_GRAPHS 256
#define IN_DIM   9
#define HID      128

typedef __attribute__((ext_vector_type(2))) float v2f;
typedef __attribute__((ext_vector_type(8))) float v8f;

// Native CDNA5 fp32 atomic add at device scope (performed at L2 atomic units,
// non-returning -> tracked with STOREcnt). Avoids a possible CAS-loop
// expansion of __hip_atomic_fetch_add.
__device__ __forceinline__ void atomic_add_f32(float* p, float v) {
    asm volatile("global_atomic_add_f32 %0, %1, off scope:SCOPE_DEV"
                 :: "v"(p), "v"(v) : "memory");
}

// ---------- degree / norm ----------
__global__ void k_fill1(float* p, int n) {
    int i = blockIdx.x * blockDim.x + threadIdx.x;
    if (i < n) p[i] = 1.0f;
}

__global__ void k_deg(const int* __restrict__ dst, float* __restrict__ deg) {
    int e = blockIdx.x * blockDim.x + threadIdx.x;
    if (e < N_EDGES) atomic_add_f32(&deg[dst[e]], 1.0f);
}

__global__ void k_rsqrt(float* p, int n) {
    int i = blockIdx.x * blockDim.x + threadIdx.x;
    if (i < n) p[i] = rsqrtf(p[i]);
}

// ---------- layer-1 GEMM: [N x 9] @ [9 x 128] ----------
__global__ void k_gemm_in(const float* __restrict__ x, const float* __restrict__ W1,
                          float* __restrict__ out) {
    int idx = blockIdx.x * blockDim.x + threadIdx.x;  // N*128
    if (idx >= N_NODES * HID) return;
    int f = idx & (HID - 1);
    int n = idx >> 7;
    const float* xr = x + n * IN_DIM;
    float acc = 0.f;
#pragma unroll
    for (int k = 0; k < IN_DIM; ++k) acc = fmaf(xr[k], W1[k * HID + f], acc);
    out[idx] = acc;
}

// ---------- layer-2/3 GEMM via V_WMMA_F32_16X16X4_F32 ----------
// One wave computes one 16x16 tile of out = (relu?)(A) @ W, A:[N x128], W:[128x128].
// A-frag (16x4 f32): lanes 0-15 hold K=k0,k0+1; lanes 16-31 hold K=k0+2,k0+3.
// B-frag (4x16 f32): mirror layout. C/D: VGPR v -> M = half*8+v, N = lane&15.
template <bool RELU>
__global__ void k_gemm_wmma(const float* __restrict__ A, const float* __restrict__ W,
                            float* __restrict__ out) {
    int wave = (blockIdx.x * blockDim.x + threadIdx.x) >> 5;
    int lane = threadIdx.x & 31;
    int ct = wave & 7;        // 8 column tiles of 16
    int rt = wave >> 3;       // 3125 row tiles of 16
    if (rt >= (N_NODES >> 4)) return;

    int half = lane >> 4;     // 0: lanes 0-15, 1: lanes 16-31
    int l16  = lane & 15;
    int row  = (rt << 4) + l16;
    int col  = (ct << 4) + l16;
    const float* arow = A + row * HID;

    v8f c = {0.f, 0.f, 0.f, 0.f, 0.f, 0.f, 0.f, 0.f};
#pragma unroll
    for (int k0 = 0; k0 < HID; k0 += 4) {
        int ka = k0 + (half << 1);
        float a0 = arow[ka];
        float a1 = arow[ka + 1];
        if (RELU) { a0 = fmaxf(a0, 0.f); a1 = fmaxf(a1, 0.f); }
        v2f a; a.x = a0; a.y = a1;
        v2f b; b.x = W[ka * HID + col]; b.y = W[(ka + 1) * HID + col];
        c = __builtin_amdgcn_wmma_f32_16x16x4_f32(false, a, false, b,
                                                  (short)0, c, false, false);
    }
    float* orow = out + ((rt << 4) + (half << 3)) * HID + (ct << 4) + l16;
#pragma unroll
    for (int v = 0; v < 8; ++v) orow[v * HID] = c[v];
}

// ---------- out = hw * dis^2 + bias (self-loop term) ----------
__global__ void k_init_out(const float* __restrict__ hw, const float* __restrict__ dis,
                           const float* __restrict__ bias, float* __restrict__ out) {
    int idx = blockIdx.x * blockDim.x + threadIdx.x;
    if (idx >= N_NODES * HID) return;
    int f = idx & (HID - 1);
    int n = idx >> 7;
    float s = dis[n];
    out[idx] = fmaf(hw[idx], s * s, bias[f]);
}

// ---------- edge scatter: out[dst] += hw[src] * dis[src]*dis[dst] ----------
// one wave32 per edge; lane moves 4 contiguous floats (global_load_b128 gather,
// 4x native global_atomic_add_f32 scatter). Edge metadata is forced wave-
// uniform via readfirstlane so it can use the scalar (SMEM) path.
__global__ void k_edge(const int* __restrict__ src, const int* __restrict__ dst,
                       const float* __restrict__ dis, const float* __restrict__ hw,
                       float* __restrict__ out) {
    int t = blockIdx.x * blockDim.x + threadIdx.x;
    int e = t >> 5;
    if (e >= N_EDGES) return;
    e = __builtin_amdgcn_readfirstlane(e);
    int lane = t & 31;
    int s = src[e];
    int d = dst[e];
    float w = dis[s] * dis[d];
    const float4 h4 = *(const float4*)(hw + s * HID + lane * 4);
    float* o = out + d * HID + lane * 4;
    atomic_add_f32(o + 0, h4.x * w);
    atomic_add_f32(o + 1, h4.y * w);
    atomic_add_f32(o + 2, h4.z * w);
    atomic_add_f32(o + 3, h4.w * w);
}

// ---------- pooling ----------
__global__ void k_pool_zero(float* __restrict__ sums, float* __restrict__ counts) {
    int i = blockIdx.x * blockDim.x + threadIdx.x;
    if (i < N_GRAPHS * HID) sums[i] = 0.f;
    if (i < N_GRAPHS) counts[i] = 0.f;
}

__global__ void k_pool_acc(const float* __restrict__ h, const int* __restrict__ batch,
                           float* __restrict__ sums, float* __restrict__ counts) {
    int t = blockIdx.x * blockDim.x + threadIdx.x;
    int n = t >> 5;
    if (n >= N_NODES) return;
    n = __builtin_amdgcn_readfirstlane(n);
    int lane = t & 31;
    int g = batch[n];
    const float4 h4 = *(const float4*)(h + n * HID + lane * 4);
    float* o = sums + g * HID + lane * 4;
    atomic_add_f32(o + 0, h4.x);
    atomic_add_f32(o + 1, h4.y);
    atomic_add_f32(o + 2, h4.z);
    atomic_add_f32(o + 3, h4.w);
    if (lane == 0) atomic_add_f32(&counts[g], 1.0f);
}

__global__ void k_pool_div(float* __restrict__ sums, const float* __restrict__ counts) {
    int i = blockIdx.x * blockDim.x + threadIdx.x;
    if (i >= N_GRAPHS * HID) return;
    int g = i >> 7;
    sums[i] = sums[i] / fmaxf(counts[g], 1.0f);
}

extern "C" void kernel_launch(void* const* d_in, const int* in_sizes, int n_in,
                              void* d_out, int out_size, void* d_ws, size_t ws_size,
                              hipStream_t stream) {
    const float* x        = (const float*)d_in[0];
    const int*   eidx     = (const int*)d_in[1];   // [2, E]
    const int*   batch    = (const int*)d_in[2];
    const float* W1       = (const float*)d_in[3];
    const float* b1       = (const float*)d_in[4];
    const float* W2       = (const float*)d_in[5];
    const float* b2       = (const float*)d_in[6];
    const float* W3       = (const float*)d_in[7];
    const float* b3       = (const float*)d_in[8];
    float* out = (float*)d_out;

    const int* src = eidx;
    const int* dst = eidx + N_EDGES;

    // workspace layout (floats)
    float* ws     = (float*)d_ws;
    float* dis    = ws;                                // N
    float* hw     = dis + N_NODES;                     // N*128 (GEMM scratch)
    float* hA     = hw + (size_t)N_NODES * HID;        // N*128 (layer in/out)
    float* counts = hA + (size_t)N_NODES * HID;        // N_GRAPHS

    const int B = 256;
    int g_nf   = (N_NODES * HID + B - 1) / B;          // 25000
    int g_n    = (N_NODES + B - 1) / B;
    int g_e    = (N_EDGES + B - 1) / B;
    int g_ew   = (N_EDGES * 32 + B - 1) / B;           // wave-per-edge
    int g_nw   = (N_NODES * 32 + B - 1) / B;           // wave-per-node
    int g_wmma = ((N_NODES >> 4) * 8 + (B >> 5) - 1) / (B >> 5);  // 3125 blocks

    // normalization
    k_fill1<<<g_n, B, 0, stream>>>(dis, N_NODES);
    k_deg<<<g_e, B, 0, stream>>>(dst, dis);
    k_rsqrt<<<g_n, B, 0, stream>>>(dis, N_NODES);

    // layer 1: x @ W1 ; relu deferred (fused into layer-2 WMMA A-load)
    k_gemm_in<<<g_nf, B, 0, stream>>>(x, W1, hw);
    k_init_out<<<g_nf, B, 0, stream>>>(hw, dis, b1, hA);
    k_edge<<<g_ew, B, 0, stream>>>(src, dst, dis, hw, hA);

    // layer 2: relu(hA) @ W2
    k_gemm_wmma<true><<<g_wmma, B, 0, stream>>>(hA, W2, hw);
    k_init_out<<<g_nf, B, 0, stream>>>(hw, dis, b2, hA);
    k_edge<<<g_ew, B, 0, stream>>>(src, dst, dis, hw, hA);

    // layer 3: relu(hA) @ W3 (no relu on output)
    k_gemm_wmma<true><<<g_wmma, B, 0, stream>>>(hA, W3, hw);
    k_init_out<<<g_nf, B, 0, stream>>>(hw, dis, b3, hA);
    k_edge<<<g_ew, B, 0, stream>>>(src, dst, dis, hw, hA);

    // mean pool
    k_pool_zero<<<(N_GRAPHS * HID + B - 1) / B, B, 0, stream>>>(out, counts);
    k_pool_acc<<<g_nw, B, 0, stream>>>(hA, batch, out, counts);
    k_pool_div<<<(N_GRAPHS * HID + B - 1) / B, B, 0, stream>>>(out, counts);
}